// GCN_55929064128859
// MI455X (gfx1250) — compile-verified
//
#include <hip/hip_runtime.h>

#define NN 40000
#define NE 640000
#define DD 128
#define NC 40
#define NG 128

typedef __attribute__((ext_vector_type(2))) float v2f;
typedef __attribute__((ext_vector_type(8))) float v8f;

// Native fp32 global atomic add. unsafeAtomicAdd is HIP's documented route to
// the hardware global_atomic_add_f32 (no CAS expansion).
__device__ __forceinline__ void atomAddF(float* p, float v) {
#if defined(__HIP_DEVICE_COMPILE__)
  unsafeAtomicAdd(p, v);
#else
  __hip_atomic_fetch_add(p, v, __ATOMIC_RELAXED, __HIP_MEMORY_SCOPE_AGENT);
#endif
}

// ---------------- degrees (integer atomics: always native) ----------------
__global__ void deg_kernel(const int* __restrict__ src, const int* __restrict__ dst,
                           unsigned* __restrict__ deg_out, unsigned* __restrict__ deg_in) {
  int e = blockIdx.x * blockDim.x + threadIdx.x;
  if (e < NE) {
    atomicAdd(&deg_out[src[e]], 1u);
    atomicAdd(&deg_in[dst[e]], 1u);
  }
}

__global__ void norm_kernel(const unsigned* __restrict__ dout, const unsigned* __restrict__ din,
                            float* __restrict__ ns, float* __restrict__ nd) {
  int i = blockIdx.x * blockDim.x + threadIdx.x;
  if (i < NN) {
    ns[i] = rsqrtf(fmaxf((float)dout[i], 1.0f));
    nd[i] = rsqrtf(fmaxf((float)din[i], 1.0f));
  }
}

// ---------------- fp32 WMMA GEMM: Y[40000,128] = (X * rowscale) @ W[128,128] ----------------
// blockDim = 256 (8 waves). blockIdx.x = 16-row strip; wave w owns column tile n0 = 16*w.
// V_WMMA_F32_16X16X4_F32: A frag float2/lane, B frag float2/lane, C/D v8f/lane.
__global__ void __launch_bounds__(256) gemm_rs_wmma(const float* __restrict__ X,
                                                    const float* __restrict__ rs,
                                                    const float* __restrict__ W,
                                                    float* __restrict__ Y) {
  const int lane = threadIdx.x & 31;
  const int wave = threadIdx.x >> 5;
  const int hl   = lane >> 4;        // 0: K pair {k,k+1}; 1: K pair {k+2,k+3}
  const int l    = lane & 15;
  const int row0 = blockIdx.x * 16;
  const int n0   = wave * 16;
  const int m    = row0 + l;
  const float scale = rs[m];
  const float* __restrict__ arow = X + (size_t)m * DD;

  v8f c = {};
#pragma unroll
  for (int k = 0; k < DD; k += 4) {
    const int ka = k + 2 * hl;
    v2f a, b;
    a.x = arow[ka]     * scale;
    a.y = arow[ka + 1] * scale;
    b.x = W[ka * DD + n0 + l];
    b.y = W[(ka + 1) * DD + n0 + l];
    c = __builtin_amdgcn_wmma_f32_16x16x4_f32(false, a, false, b, (short)0, c, false, false);
  }
#pragma unroll
  for (int r = 0; r < 8; ++r) {
    Y[(size_t)(row0 + 8 * hl + r) * DD + n0 + l] = c[r];
  }
}

// ---------------- SpMM: AGG[dst] += H[src], one wave per edge ----------------
__global__ void spmm_kernel(const float* __restrict__ H, const int* __restrict__ src,
                            const int* __restrict__ dst, float* __restrict__ AGG) {
  const int lane = threadIdx.x & 31;
  const int e = blockIdx.x * (blockDim.x >> 5) + (threadIdx.x >> 5);
  if (e >= NE) return;
  const int s = src[e];
  const int d = dst[e];
  const float4 v = *(const float4*)(H + (size_t)s * DD + lane * 4);
  float* out = AGG + (size_t)d * DD + lane * 4;
  atomAddF(out + 0, v.x);
  atomAddF(out + 1, v.y);
  atomAddF(out + 2, v.z);
  atomAddF(out + 3, v.w);
}

// ---------------- h = relu(agg * norm_dst + b) ----------------
__global__ void post_relu(const float* __restrict__ AGG, const float* __restrict__ nd,
                          const float* __restrict__ bias, float* __restrict__ OUT) {
  int idx = blockIdx.x * blockDim.x + threadIdx.x;
  if (idx < NN * DD) {
    int node = idx >> 7, dcol = idx & (DD - 1);
    OUT[idx] = fmaxf(AGG[idx] * nd[node] + bias[dcol], 0.0f);
  }
}

// ---------------- h2 = agg * norm_dst + b; pooled sums += h2 ----------------
__global__ void post_pool(const float* __restrict__ AGG, const float* __restrict__ nd,
                          const float* __restrict__ bias, const int* __restrict__ gid,
                          float* __restrict__ sums) {
  int idx = blockIdx.x * blockDim.x + threadIdx.x;
  if (idx < NN * DD) {
    int node = idx >> 7, dcol = idx & (DD - 1);
    float v = AGG[idx] * nd[node] + bias[dcol];
    atomAddF(&sums[gid[node] * DD + dcol], v);
  }
}

__global__ void count_kernel(const int* __restrict__ gid, float* __restrict__ cnt) {
  int i = blockIdx.x * blockDim.x + threadIdx.x;
  if (i < NN) atomAddF(&cnt[gid[i]], 1.0f);
}

// ---------------- out[128,40] = (sums/cnt) @ Wc + bc ----------------
__global__ void classify(const float* __restrict__ sums, const float* __restrict__ cnt,
                         const float* __restrict__ Wc, const float* __restrict__ bc,
                         float* __restrict__ out) {
  int idx = blockIdx.x * blockDim.x + threadIdx.x;
  if (idx < NG * NC) {
    int g = idx / NC, cc = idx % NC;
    float inv = 1.0f / fmaxf(cnt[g], 1.0f);
    float acc = bc[cc];
#pragma unroll 8
    for (int k = 0; k < DD; ++k)
      acc += (sums[g * DD + k] * inv) * Wc[k * NC + cc];
    out[idx] = acc;
  }
}

extern "C" void kernel_launch(void* const* d_in, const int* in_sizes, int n_in,
                              void* d_out, int out_size, void* d_ws, size_t ws_size,
                              hipStream_t stream) {
  const float* feat = (const float*)d_in[0];
  const float* W1   = (const float*)d_in[1];
  const float* b1   = (const float*)d_in[2];
  const float* W2   = (const float*)d_in[3];
  const float* b2   = (const float*)d_in[4];
  const float* Wc   = (const float*)d_in[5];
  const float* bc   = (const float*)d_in[6];
  const int*   src  = (const int*)d_in[7];
  const int*   dst  = (const int*)d_in[8];
  const int*   gid  = (const int*)d_in[9];
  float* out = (float*)d_out;

  // workspace layout (floats)
  float*    ws      = (float*)d_ws;
  unsigned* deg_out = (unsigned*)ws;           // 40000
  unsigned* deg_in  = deg_out + NN;            // 40000
  float*    nsrc    = ws + 2 * NN;             // 40000
  float*    ndst    = nsrc + NN;               // 40000
  float*    bufA    = ndst + NN;               // 40000*128
  float*    bufB    = bufA + (size_t)NN * DD;  // 40000*128
  float*    sums    = bufB + (size_t)NN * DD;  // 128*128
  float*    cnt     = sums + NG * DD;          // 128

  const int T = 256;

  hipMemsetAsync(deg_out, 0, 2 * NN * sizeof(unsigned), stream);
  hipMemsetAsync(sums, 0, (NG * DD + NG) * sizeof(float), stream);

  deg_kernel<<<(NE + T - 1) / T, T, 0, stream>>>(src, dst, deg_out, deg_in);
  norm_kernel<<<(NN + T - 1) / T, T, 0, stream>>>(deg_out, deg_in, nsrc, ndst);

  // Layer 1
  gemm_rs_wmma<<<NN / 16, 256, 0, stream>>>(feat, nsrc, W1, bufA);
  hipMemsetAsync(bufB, 0, (size_t)NN * DD * sizeof(float), stream);
  spmm_kernel<<<NE / 8, 256, 0, stream>>>(bufA, src, dst, bufB);
  post_relu<<<(NN * DD + T - 1) / T, T, 0, stream>>>(bufB, ndst, b1, bufA);

  // Layer 2
  gemm_rs_wmma<<<NN / 16, 256, 0, stream>>>(bufA, nsrc, W2, bufB);
  hipMemsetAsync(bufA, 0, (size_t)NN * DD * sizeof(float), stream);
  spmm_kernel<<<NE / 8, 256, 0, stream>>>(bufB, src, dst, bufA);

  // Readout
  post_pool<<<(NN * DD + T - 1) / T, T, 0, stream>>>(bufA, ndst, b2, gid, sums);
  count_kernel<<<(NN + T - 1) / T, T, 0, stream>>>(gid, cnt);
  classify<<<(NG * NC + T - 1) / T, T, 0, stream>>>(sums, cnt, Wc, bc, out);
}